// GAT1_83897891160312
// MI455X (gfx1250) — compile-verified
//
#include <hip/hip_runtime.h>
#include <math.h>

// ---------------------------------------------------------------------------
// GAT layer for MI455X (gfx1250, wave32).
//   feat = x@W via V_WMMA_F32_16X16X4_F32 (fp32-exact matrix path)
//     - W prepacked once into per-lane B-fragment order -> b64 loads in loop
//     - one wave = full 16(M) x 128(N) tile -> x streamed exactly once
//   edge softmax via native float atomics (int-trick max, atomic_add_f32)
//   aggregation: 1 wave per edge, lane == output dim (wave32 == OUT_DIM)
// Working set (~58 MB ws + 12.8 MB out) is resident in the 192 MB L2.
// ---------------------------------------------------------------------------

typedef float v2f __attribute__((ext_vector_type(2)));
typedef float v8f __attribute__((ext_vector_type(8)));

#define N_NODES   100000
#define N_EDGES   1600000
#define IN_DIM    256
#define HEADS     4
#define OUT_DIM   32
#define FEAT_DIM  128          // HEADS*OUT_DIM
#define NEG_SLOPE 0.2f
#define KSTEPS    (IN_DIM / 4) // 64
#define NTILES    (FEAT_DIM / 16) // 8

// ---------------------------------------------------------------------------
// Kernel 0: prepack W[256,128] into WMMA B-fragment order.
// Wp[((kstep*8 + tile)*32 + lane)*2 + {0,1}] = W[(kstep*4 + kb + {0,1})*128 + n]
//   kb = 2*(lane>>4), n = tile*16 + (lane&15)
// 16384 threads, 128 KB total — negligible cost, makes the GEMM inner loop
// load each B fragment as one coalesced global_load_b64.
// ---------------------------------------------------------------------------
__global__ void gat_prepack_W(const float* __restrict__ W, float* __restrict__ Wp) {
  const int t = blockIdx.x * blockDim.x + threadIdx.x;
  if (t >= KSTEPS * NTILES * 32) return;
  const int lane  = t & 31;
  const int tile  = (t >> 5) & 7;
  const int kstep = t >> 8;
  const int kb = (lane >> 4) * 2;
  const int n  = tile * 16 + (lane & 15);
  const int k  = kstep * 4 + kb;
  Wp[2 * t + 0] = W[(size_t)k * FEAT_DIM + n];
  Wp[2 * t + 1] = W[(size_t)(k + 1) * FEAT_DIM + n];
}

// ---------------------------------------------------------------------------
// Kernel 1: feat[N,128] = x[N,256] @ W[256,128] with fp32 WMMA.
// One wave computes the full 16(M) x 128(N) slab; K loops 0..255 in steps of 4.
// A-frag (16x4 f32): lane l holds row m=l&15, K = 2*(l>>4) + {0,1}
// B-frag (4x16 f32): prepacked, one b64 per fragment
// C/D   (16x16 f32): lane l, vgpr i -> M = i + 8*(l>>4), N = l&15
// ---------------------------------------------------------------------------
__global__ __launch_bounds__(256)
void gat_gemm_wmma(const float* __restrict__ x, const float* __restrict__ Wp,
                   float* __restrict__ feat) {
  const int lane = threadIdx.x & 31;
  const int mt   = blockIdx.x * 8 + (threadIdx.x >> 5);
  if (mt >= N_NODES / 16) return;                // wave-uniform exit

  const int kb = (lane >> 4) * 2;                // 0 or 2
  const float* aptr = x + (size_t)(mt * 16 + (lane & 15)) * IN_DIM + kb;
  const float* bptr = Wp + (size_t)lane * 2;     // + (kstep*8 + tile)*64

  v8f c[NTILES] = {};

  for (int ks = 0; ks < KSTEPS; ++ks) {
    const v2f a = *(const v2f*)(aptr + ks * 4);  // 8B aligned
#pragma unroll
    for (int t = 0; t < NTILES; ++t) {
      const v2f b = *(const v2f*)(bptr + (size_t)(ks * NTILES + t) * 64);
      c[t] = __builtin_amdgcn_wmma_f32_16x16x4_f32(false, a, false, b,
                                                   (short)0, c[t], false, false);
    }
  }

  const int mbase = mt * 16 + 8 * (lane >> 4);
  float* o = feat + (size_t)mbase * FEAT_DIM + (lane & 15);
#pragma unroll
  for (int t = 0; t < NTILES; ++t)
#pragma unroll
    for (int i = 0; i < 8; ++i)
      o[(size_t)i * FEAT_DIM + t * 16] = c[t][i];
}

// ---------------------------------------------------------------------------
// Kernel 2: init m = -inf, denom = 0, out = 0 (atomics accumulate each call).
// ---------------------------------------------------------------------------
__global__ void gat_init(float* __restrict__ m, float* __restrict__ denom,
                         float* __restrict__ out) {
  int t = blockIdx.x * blockDim.x + threadIdx.x;
  if (t < N_NODES * OUT_DIM) out[t] = 0.0f;
  if (t < N_NODES * HEADS) { m[t] = -INFINITY; denom[t] = 0.0f; }
}

// ---------------------------------------------------------------------------
// Kernel 3: per-(node,head) attention logits el/er (dot of 32-d head slice).
// ---------------------------------------------------------------------------
__global__ void gat_el_er(const float* __restrict__ feat,
                          const float* __restrict__ attn_l,
                          const float* __restrict__ attn_r,
                          float* __restrict__ el, float* __restrict__ er) {
  int t = blockIdx.x * blockDim.x + threadIdx.x;
  if (t >= N_NODES * HEADS) return;
  const int h = t & 3;
  const float* f  = feat + (size_t)(t >> 2) * FEAT_DIM + h * OUT_DIM;
  const float* al = attn_l + h * OUT_DIM;
  const float* ar = attn_r + h * OUT_DIM;
  float sl = 0.0f, sr = 0.0f;
#pragma unroll
  for (int d = 0; d < OUT_DIM; ++d) { float v = f[d]; sl += v * al[d]; sr += v * ar[d]; }
  el[t] = sl;
  er[t] = sr;
}

__device__ __forceinline__ float lrelu(float v) {
  return v > 0.0f ? v : NEG_SLOPE * v;
}

// Bit-trick float atomic max (native int atomics, no CAS loop). m init = -inf.
__device__ __forceinline__ void atomic_max_f32(float* addr, float v) {
  if (v >= 0.0f) atomicMax((int*)addr, __float_as_int(v));
  else           atomicMin((unsigned int*)addr, __float_as_uint(v));
}

// ---------------------------------------------------------------------------
// Kernel 4: segment max of e = lrelu(el[src]+er[dst]) into m[dst].
// ---------------------------------------------------------------------------
__global__ void gat_edge_max(const int* __restrict__ src, const int* __restrict__ dst,
                             const float* __restrict__ el, const float* __restrict__ er,
                             float* __restrict__ m) {
  int e = blockIdx.x * blockDim.x + threadIdx.x;
  if (e >= N_EDGES) return;
  const int s = src[e], d = dst[e];
  const float4 L = *(const float4*)(el + (size_t)s * HEADS);
  const float4 R = *(const float4*)(er + (size_t)d * HEADS);
  atomic_max_f32(m + (size_t)d * HEADS + 0, lrelu(L.x + R.x));
  atomic_max_f32(m + (size_t)d * HEADS + 1, lrelu(L.y + R.y));
  atomic_max_f32(m + (size_t)d * HEADS + 2, lrelu(L.z + R.z));
  atomic_max_f32(m + (size_t)d * HEADS + 3, lrelu(L.w + R.w));
}

// ---------------------------------------------------------------------------
// Kernel 5: denom[dst] += exp(e - m[dst])   (GLOBAL_ATOMIC_ADD_F32)
// ---------------------------------------------------------------------------
__global__ void gat_edge_sum(const int* __restrict__ src, const int* __restrict__ dst,
                             const float* __restrict__ el, const float* __restrict__ er,
                             const float* __restrict__ m, float* __restrict__ denom) {
  int e = blockIdx.x * blockDim.x + threadIdx.x;
  if (e >= N_EDGES) return;
  const int s = src[e], d = dst[e];
  const float4 L = *(const float4*)(el + (size_t)s * HEADS);
  const float4 R = *(const float4*)(er + (size_t)d * HEADS);
  const float4 M = *(const float4*)(m  + (size_t)d * HEADS);
  atomicAdd(denom + (size_t)d * HEADS + 0, __expf(lrelu(L.x + R.x) - M.x));
  atomicAdd(denom + (size_t)d * HEADS + 1, __expf(lrelu(L.y + R.y) - M.y));
  atomicAdd(denom + (size_t)d * HEADS + 2, __expf(lrelu(L.z + R.z) - M.z));
  atomicAdd(denom + (size_t)d * HEADS + 3, __expf(lrelu(L.w + R.w) - M.w));
}

// ---------------------------------------------------------------------------
// Kernel 6: aggregation. One wave32 per edge; lane = output dim d (0..31).
//   out[dst][d] += 0.25 * sum_h a[e,h] * feat[src][h*32+d]
// feat gathers are 4x128B coalesced bursts; the atomic burst is 128B coalesced.
// All of feat/el/er/m/denom sit in L2 (192 MB).
// ---------------------------------------------------------------------------
__global__ __launch_bounds__(256)
void gat_aggregate(const int* __restrict__ src, const int* __restrict__ dst,
                   const float* __restrict__ el, const float* __restrict__ er,
                   const float* __restrict__ m, const float* __restrict__ denom,
                   const float* __restrict__ feat, float* __restrict__ out) {
  const int lane = threadIdx.x & 31;
  const int e = blockIdx.x * 8 + (threadIdx.x >> 5);
  if (e >= N_EDGES) return;
  const int s = src[e], d = dst[e];
  const float4 L  = *(const float4*)(el    + (size_t)s * HEADS);
  const float4 R  = *(const float4*)(er    + (size_t)d * HEADS);
  const float4 M  = *(const float4*)(m     + (size_t)d * HEADS);
  const float4 Dn = *(const float4*)(denom + (size_t)d * HEADS);
  const float a0 = __expf(lrelu(L.x + R.x) - M.x) / fmaxf(Dn.x, 1e-9f);
  const float a1 = __expf(lrelu(L.y + R.y) - M.y) / fmaxf(Dn.y, 1e-9f);
  const float a2 = __expf(lrelu(L.z + R.z) - M.z) / fmaxf(Dn.z, 1e-9f);
  const float a3 = __expf(lrelu(L.w + R.w) - M.w) / fmaxf(Dn.w, 1e-9f);

  const float* f = feat + (size_t)s * FEAT_DIM + lane;
  const float acc = 0.25f * (a0 * f[0] + a1 * f[32] + a2 * f[64] + a3 * f[96]);
  atomicAdd(out + (size_t)d * OUT_DIM + lane, acc);
}

// ---------------------------------------------------------------------------
extern "C" void kernel_launch(void* const* d_in, const int* in_sizes, int n_in,
                              void* d_out, int out_size, void* d_ws, size_t ws_size,
                              hipStream_t stream) {
  const float* x      = (const float*)d_in[0];
  const float* W      = (const float*)d_in[1];
  const float* attn_l = (const float*)d_in[2];
  const float* attn_r = (const float*)d_in[3];
  const int*   src    = (const int*)d_in[4];
  const int*   dst    = (const int*)d_in[5];
  float* out = (float*)d_out;

  // Workspace layout (floats): feat | el | er | m | denom | Wp  (~57.7 MB)
  float* ws    = (float*)d_ws;
  float* feat  = ws;
  float* el    = feat + (size_t)N_NODES * FEAT_DIM;
  float* er    = el   + (size_t)N_NODES * HEADS;
  float* mbuf  = er   + (size_t)N_NODES * HEADS;
  float* denom = mbuf + (size_t)N_NODES * HEADS;
  float* Wp    = denom + (size_t)N_NODES * HEADS;   // 32768 floats (128 KB)

  gat_prepack_W<<<(KSTEPS * NTILES * 32 + 255) / 256, 256, 0, stream>>>(W, Wp);
  const int MT = N_NODES / 16;                       // 6250 wave-tiles
  gat_gemm_wmma<<<(MT + 7) / 8, 256, 0, stream>>>(x, Wp, feat);
  gat_init<<<(N_NODES * OUT_DIM + 255) / 256, 256, 0, stream>>>(mbuf, denom, out);
  gat_el_er<<<(N_NODES * HEADS + 255) / 256, 256, 0, stream>>>(feat, attn_l, attn_r, el, er);
  gat_edge_max<<<(N_EDGES + 255) / 256, 256, 0, stream>>>(src, dst, el, er, mbuf);
  gat_edge_sum<<<(N_EDGES + 255) / 256, 256, 0, stream>>>(src, dst, el, er, mbuf, denom);
  gat_aggregate<<<(N_EDGES + 7) / 8, 256, 0, stream>>>(src, dst, el, er, mbuf, denom, feat, out);
}